// KGMTRS_12773232738836
// MI455X (gfx1250) — compile-verified
//
#include <hip/hip_runtime.h>

typedef __attribute__((ext_vector_type(2))) float v2f;
typedef __attribute__((ext_vector_type(8))) float v8f;

#define NGRID 50000
#define NCAT  5000
#define DIM   128
#define ODIM  128
#define EEDGE 600000
#define BATCH 1024

#define CAP_GRID 2048   // max distinct needed grid nodes (pos+neg)
#define CAP_CAT  1024   // max distinct needed cat nodes

// ---------------- workspace layout (bytes) ----------------
// slot_grid : int[NGRID]           @ 0
// slot_cat  : int[NCAT]            @ 200000
// counters  : int[4]               @ 220000   {grid_slots, cat_slots, cnt_c2g, cnt_g2c}
// list_c2g  : int[EEDGE]           @ 220016
// list_g2c  : int[EEDGE]           @ 2620016
// nh_grid   : float[CAP_GRID*128]  @ 5020016
// nh_cat    : float[CAP_CAT*128]   @ 6068592
// total ~6.6 MB

__global__ void init_ws(int* slot_grid, int* slot_cat, int* counters,
                        float* nh_grid, float* nh_cat) {
    int i = blockIdx.x * blockDim.x + threadIdx.x;
    int stride = gridDim.x * blockDim.x;
    for (int k = i; k < NGRID; k += stride) slot_grid[k] = -1;
    for (int k = i; k < NCAT;  k += stride) slot_cat[k]  = -1;
    if (i < 4) counters[i] = 0;
    for (int k = i; k < CAP_GRID * DIM; k += stride) nh_grid[k] = 0.0f;
    for (int k = i; k < CAP_CAT  * DIM; k += stride) nh_cat[k]  = 0.0f;
}

__global__ void mark_nodes(const int* __restrict__ cat_ids,
                           const int* __restrict__ pos_ids,
                           const int* __restrict__ neg_ids,
                           int* slot_grid, int* slot_cat) {
    int i = blockIdx.x * blockDim.x + threadIdx.x;
    if (i < BATCH) {
        slot_cat[cat_ids[i]]  = -2;   // idempotent marks
        slot_grid[pos_ids[i]] = -2;
        slot_grid[neg_ids[i]] = -2;
    }
}

__global__ void assign_slots(int* slots, int n, int* counter) {
    int i = blockIdx.x * blockDim.x + threadIdx.x;
    if (i < n && slots[i] == -2) slots[i] = atomicAdd(counter, 1);
}

__global__ void filter_edges(const int* __restrict__ dst,
                             const int* __restrict__ slots,
                             int* list, int* count) {
    int e = blockIdx.x * blockDim.x + threadIdx.x;
    if (e < EEDGE && slots[dst[e]] >= 0) {
        int idx = atomicAdd(count, 1);
        list[idx] = e;
    }
}

// one wave (32 lanes) per matched edge; lane owns 4 consecutive floats
__global__ void scatter_edges(const int* __restrict__ list,
                              const int* __restrict__ count,
                              const int* __restrict__ src,
                              const int* __restrict__ dst,
                              const float* __restrict__ att,
                              const float* __restrict__ vsrc,
                              const int* __restrict__ slots,
                              float* nh) {
    const int lane   = threadIdx.x & 31;
    const int wave   = (blockIdx.x * blockDim.x + threadIdx.x) >> 5;
    const int nwaves = (gridDim.x * blockDim.x) >> 5;
    const int n = count[0];
    for (int idx = wave; idx < n; idx += nwaves) {
        const int   e = list[idx];
        const int   s = src[e];
        const int   d = dst[e];
        const float a = att[e];
        const float4 v = ((const float4*)(vsrc + (size_t)s * DIM))[lane];
        float* base = nh + (size_t)slots[d] * DIM + lane * 4;
        atomicAdd(base + 0, v.x * a);
        atomicAdd(base + 1, v.y * a);
        atomicAdd(base + 2, v.z * a);
        atomicAdd(base + 3, v.w * a);
    }
}

// ---------------------------------------------------------------------------
// dual-branch GEMM + LeakyReLU + gather.
// Block = 64 threads (2 waves). Each wave owns 16 output rows.
// W1 column tiles are cooperatively staged into LDS *transposed* so a lane's
// B fragment (two consecutive K values at fixed N) is one ds_load_b64.
// A fragments are preloaded to registers (tile-invariant).
// K accumulation is split into two independent WMMA chains per branch.
// ---------------------------------------------------------------------------
#define XP  132   // x-tile pitch (dwords): conflict-free column reads, float4 aligned
#define WPK 132   // transposed W-tile pitch (dwords)

__global__ __launch_bounds__(64)
void dual_gemm_gather(const float* __restrict__ v_grid,
                      const float* __restrict__ v_cat,
                      const float* __restrict__ W1,
                      const float* __restrict__ b1,
                      const int* __restrict__ cat_ids,
                      const int* __restrict__ pos_ids,
                      const int* __restrict__ neg_ids,
                      const int* __restrict__ slot_grid,
                      const int* __restrict__ slot_cat,
                      const float* __restrict__ nh_grid,
                      const float* __restrict__ nh_cat,
                      float* __restrict__ out) {
    __shared__ float xtiles[2][2][16 * XP];   // [wave][add|mul][16 x 128 padded]
    __shared__ float wtile[16 * WPK];         // transposed: wtile[nn*WPK + k]

    const int tid  = threadIdx.x;
    const int wave = tid >> 5;
    const int lane = tid & 31;
    const int r0   = (blockIdx.x * 2 + wave) * 16;   // 16 rows per wave

    // segment select (segments are 1024-aligned; 32-row blocks never straddle)
    const int*   ids;  const float* vtab; const float* nhtab; const int* slots; int ibase;
    if (r0 < BATCH)          { ids = cat_ids; vtab = v_cat;  nhtab = nh_cat;  slots = slot_cat;  ibase = r0; }
    else if (r0 < 2 * BATCH) { ids = pos_ids; vtab = v_grid; nhtab = nh_grid; slots = slot_grid; ibase = r0 - BATCH; }
    else                     { ids = neg_ids; vtab = v_grid; nhtab = nh_grid; slots = slot_grid; ibase = r0 - 2 * BATCH; }

    float* xadd = &xtiles[wave][0][0];
    float* xmul = &xtiles[wave][1][0];

    // stage v+nh and v*nh tiles (16 x 128) into LDS (wave-private)
    for (int m = 0; m < 16; ++m) {
        const int node = ids[ibase + m];
        const float4 v4 = ((const float4*)(vtab  + (size_t)node * DIM))[lane];
        const int    sl = slots[node];
        const float4 n4 = ((const float4*)(nhtab + (size_t)sl   * DIM))[lane];
        float4 xa, xm;
        xa.x = v4.x + n4.x; xa.y = v4.y + n4.y; xa.z = v4.z + n4.z; xa.w = v4.w + n4.w;
        xm.x = v4.x * n4.x; xm.y = v4.y * n4.y; xm.z = v4.z * n4.z; xm.w = v4.w * n4.w;
        *((float4*)&xadd[m * XP + lane * 4]) = xa;
        *((float4*)&xmul[m * XP + lane * 4]) = xm;
    }
    __syncthreads();

    // WMMA fragment roles (16x16x4 f32, wave32):
    //   A: lane -> M = lane%16, holds K = 2*(lane/16) + {0,1}
    //   B: lane -> N = lane%16, holds K = 2*(lane/16) + {0,1}
    //   C/D: lane -> N = lane%16, VGPR v -> M = v + 8*(lane/16)
    const int lm   = lane & 15;
    const int lh   = lane >> 4;
    const int koff = lh * 2;

    // preload A fragments for all 32 K-steps (contiguous, 8B-aligned -> ds_load_b64)
    v2f Aa[32], Am[32];
    #pragma unroll
    for (int s = 0; s < 32; ++s) {
        const int kb = s * 4 + koff;
        Aa[s] = *(const v2f*)&xadd[lm * XP + kb];
        Am[s] = *(const v2f*)&xmul[lm * XP + kb];
    }

    for (int nt = 0; nt < 8; ++nt) {
        const int n0 = nt * 16;

        // cooperatively stage W1[:, n0:n0+16] transposed into LDS
        #pragma unroll
        for (int it = 0; it < 8; ++it) {
            const int idx = it * 64 + tid;          // 0..511
            const int k   = idx >> 2;               // 0..127
            const int j4  = (idx & 3) << 2;         // 0,4,8,12
            const float4 w = *(const float4*)&W1[(size_t)k * ODIM + n0 + j4];
            wtile[(j4 + 0) * WPK + k] = w.x;
            wtile[(j4 + 1) * WPK + k] = w.y;
            wtile[(j4 + 2) * WPK + k] = w.z;
            wtile[(j4 + 3) * WPK + k] = w.w;
        }
        __syncthreads();

        const float bias = b1[n0 + lm];
        v8f ca0, ca1, cm0, cm1;
        #pragma unroll
        for (int i = 0; i < 8; ++i) { ca0[i] = bias; cm0[i] = bias; ca1[i] = 0.0f; cm1[i] = 0.0f; }

        #pragma unroll
        for (int s = 0; s < 32; s += 2) {
            const v2f b0 = *(const v2f*)&wtile[lm * WPK + (s * 4 + koff)];
            const v2f b1v = *(const v2f*)&wtile[lm * WPK + ((s + 1) * 4 + koff)];
            ca0 = __builtin_amdgcn_wmma_f32_16x16x4_f32(false, Aa[s],     false, b0,
                                                        (short)0, ca0, false, false);
            cm0 = __builtin_amdgcn_wmma_f32_16x16x4_f32(false, Am[s],     false, b0,
                                                        (short)0, cm0, false, false);
            ca1 = __builtin_amdgcn_wmma_f32_16x16x4_f32(false, Aa[s + 1], false, b1v,
                                                        (short)0, ca1, false, false);
            cm1 = __builtin_amdgcn_wmma_f32_16x16x4_f32(false, Am[s + 1], false, b1v,
                                                        (short)0, cm1, false, false);
        }

        #pragma unroll
        for (int i = 0; i < 8; ++i) {
            float va = ca0[i] + ca1[i];
            float vm = cm0[i] + cm1[i];
            va = va > 0.0f ? va : 0.01f * va;   // leaky_relu, slope 0.01
            vm = vm > 0.0f ? vm : 0.01f * vm;
            out[(size_t)(r0 + i + lh * 8) * ODIM + n0 + lm] = va + vm;
        }
        __syncthreads();   // wtile will be restaged next iteration
    }
}

extern "C" void kernel_launch(void* const* d_in, const int* in_sizes, int n_in,
                              void* d_out, int out_size, void* d_ws, size_t ws_size,
                              hipStream_t stream) {
    (void)in_sizes; (void)n_in; (void)out_size; (void)ws_size;

    const float* v_grid   = (const float*)d_in[0];
    const float* v_cat    = (const float*)d_in[1];
    const float* att_c2g  = (const float*)d_in[2];
    const float* att_g2c  = (const float*)d_in[3];
    const float* W1       = (const float*)d_in[4];
    const float* b1       = (const float*)d_in[5];
    const int*   src_c2g  = (const int*)d_in[6];
    const int*   dst_c2g  = (const int*)d_in[7];
    const int*   src_g2c  = (const int*)d_in[8];
    const int*   dst_g2c  = (const int*)d_in[9];
    const int*   cat_ids  = (const int*)d_in[10];
    const int*   pos_ids  = (const int*)d_in[11];
    const int*   neg_ids  = (const int*)d_in[12];
    float* out = (float*)d_out;

    char* ws = (char*)d_ws;
    int*   slot_grid = (int*)(ws + 0);
    int*   slot_cat  = (int*)(ws + 200000);
    int*   counters  = (int*)(ws + 220000);
    int*   list_c2g  = (int*)(ws + 220016);
    int*   list_g2c  = (int*)(ws + 2620016);
    float* nh_grid   = (float*)(ws + 5020016);
    float* nh_cat    = (float*)(ws + 6068592);

    // 1) init slots / counters / compact nh
    init_ws<<<1536, 256, 0, stream>>>(slot_grid, slot_cat, counters, nh_grid, nh_cat);

    // 2) mark needed nodes
    mark_nodes<<<(BATCH + 255) / 256, 256, 0, stream>>>(cat_ids, pos_ids, neg_ids,
                                                        slot_grid, slot_cat);

    // 3) assign compact slots
    assign_slots<<<(NGRID + 255) / 256, 256, 0, stream>>>(slot_grid, NGRID, counters + 0);
    assign_slots<<<(NCAT  + 255) / 256, 256, 0, stream>>>(slot_cat,  NCAT,  counters + 1);

    // 4) filter edges whose destination is needed
    filter_edges<<<(EEDGE + 255) / 256, 256, 0, stream>>>(dst_c2g, slot_grid,
                                                          list_c2g, counters + 2);
    filter_edges<<<(EEDGE + 255) / 256, 256, 0, stream>>>(dst_g2c, slot_cat,
                                                          list_g2c, counters + 3);

    // 5) scatter matched edges into compact nh (one wave per edge)
    scatter_edges<<<512, 256, 0, stream>>>(list_c2g, counters + 2, src_c2g, dst_c2g,
                                           att_c2g, v_cat,  slot_grid, nh_grid);
    scatter_edges<<<512, 256, 0, stream>>>(list_g2c, counters + 3, src_g2c, dst_g2c,
                                           att_g2c, v_grid, slot_cat,  nh_cat);

    // 6) dual-branch GEMM + LeakyReLU + gather, fp32 WMMA, 32 rows per block
    dual_gemm_gather<<<(3 * BATCH) / 32, 64, 0, stream>>>(
        v_grid, v_cat, W1, b1, cat_ids, pos_ids, neg_ids,
        slot_grid, slot_cat, nh_grid, nh_cat, out);
}